// KLDiscretLoss_Joints_88776974008376
// MI455X (gfx1250) — compile-verified
//
#include <hip/hip_runtime.h>
#include <hip/hip_bf16.h>
#include <math.h>

// Problem constants (from reference: B=256, J=17, D=2, BINS=2048)
#define NB   256
#define NJ   17
#define ND   2
#define BINS 2048
#define ROWS (NB * NJ * ND)   // 8704 rows of 2048 f32
#define TPB  256              // 8 wave32s per block; 8 elems/thread/tensor

typedef float v4f __attribute__((ext_vector_type(4)));
typedef int   v4i __attribute__((ext_vector_type(4)));

// ---- CDNA5 async global->LDS path (guarded; falls back to NT register loads)
#if defined(__HIP_DEVICE_COMPILE__) && defined(__gfx1250__) && \
    __has_builtin(__builtin_amdgcn_global_load_async_to_lds_b128)
#define USE_ASYNC_LDS 1
typedef __attribute__((address_space(1))) v4i* as1_v4i;
typedef __attribute__((address_space(3))) v4i* as3_v4i;
__device__ __forceinline__ void wait_asynccnt0() {
#if __has_builtin(__builtin_amdgcn_s_wait_asynccnt)
  __builtin_amdgcn_s_wait_asynccnt(0);
#else
  asm volatile("s_wait_asynccnt 0" ::: "memory");
#endif
}
#else
#define USE_ASYNC_LDS 0
#endif

// Pairwise block reduction over 256 threads (8 wave32s).
// do_max=true -> componentwise max, else sum. Result broadcast to all threads.
__device__ __forceinline__ float2 breduce2(float a, float b, bool do_max,
                                           float* sA, float* sB, float* bc,
                                           int lane, int wave) {
#pragma unroll
  for (int m = 16; m >= 1; m >>= 1) {
    float a2 = __shfl_xor(a, m, 32);
    float b2 = __shfl_xor(b, m, 32);
    if (do_max) { a = fmaxf(a, a2); b = fmaxf(b, b2); }
    else        { a += a2;          b += b2; }
  }
  if (lane == 0) { sA[wave] = a; sB[wave] = b; }
  __syncthreads();
  if (wave == 0) {
    float idn = do_max ? -3.402823466e38f : 0.0f;
    float ra = (lane < 8) ? sA[lane] : idn;
    float rb = (lane < 8) ? sB[lane] : idn;
#pragma unroll
    for (int m = 4; m >= 1; m >>= 1) {
      float a2 = __shfl_xor(ra, m, 32);
      float b2 = __shfl_xor(rb, m, 32);
      if (do_max) { ra = fmaxf(ra, a2); rb = fmaxf(rb, b2); }
      else        { ra += a2;           rb += b2; }
    }
    if (lane == 0) { bc[0] = ra; bc[1] = rb; }
  }
  __syncthreads();
  float2 r; r.x = bc[0]; r.y = bc[1];
  __syncthreads();   // protect sA/sB/bc reuse on the next call
  return r;
}

// Kernel 1: one block per (b,j,d) row. Computes mean-over-bins KL for the row.
__global__ __launch_bounds__(TPB)
void kl_rows_kernel(const float* __restrict__ pred,
                    const float* __restrict__ tgt,
                    float* __restrict__ row_out) {
  const int row  = blockIdx.x;
  const int tid  = threadIdx.x;
  const int lane = tid & 31;
  const int wave = tid >> 5;
  const size_t base = (size_t)row * BINS;

  __shared__ float sA[8], sB[8], bc[2];

  // Each thread owns 16B chunks (tid) and (tid+256) of the 512-chunk row,
  // i.e. floats [tid*4, tid*4+4) and [1024+tid*4, ...). Fully coalesced.
  float o[8], t[8];

#if USE_ASYNC_LDS
  __shared__ float ldsO[BINS];
  __shared__ float ldsT[BINS];
  {
    v4i* go = (v4i*)(pred + base);   // C-style cast drops const
    v4i* gt = (v4i*)(tgt  + base);
    v4i* lo = (v4i*)ldsO;
    v4i* lt = (v4i*)ldsT;
    __builtin_amdgcn_global_load_async_to_lds_b128(
        (as1_v4i)(go + tid),       (as3_v4i)(lo + tid),       0, 0);
    __builtin_amdgcn_global_load_async_to_lds_b128(
        (as1_v4i)(go + tid + TPB), (as3_v4i)(lo + tid + TPB), 0, 0);
    __builtin_amdgcn_global_load_async_to_lds_b128(
        (as1_v4i)(gt + tid),       (as3_v4i)(lt + tid),       0, 0);
    __builtin_amdgcn_global_load_async_to_lds_b128(
        (as1_v4i)(gt + tid + TPB), (as3_v4i)(lt + tid + TPB), 0, 0);
    // Each thread reads back exactly the chunks its own wave issued, so a
    // per-wave asynccnt wait is sufficient (no block barrier required here).
    wait_asynccnt0();
#pragma unroll
    for (int k = 0; k < 4; ++k) {
      o[k]     = ldsO[tid * 4 + k];
      o[k + 4] = ldsO[tid * 4 + 1024 + k];
      t[k]     = ldsT[tid * 4 + k];
      t[k + 4] = ldsT[tid * 4 + 1024 + k];
    }
  }
#else
  {
    const v4f* op = (const v4f*)(pred + base);
    const v4f* tp = (const v4f*)(tgt  + base);
    v4f o0 = __builtin_nontemporal_load(op + tid);
    v4f o1 = __builtin_nontemporal_load(op + tid + TPB);
    v4f t0 = __builtin_nontemporal_load(tp + tid);
    v4f t1 = __builtin_nontemporal_load(tp + tid + TPB);
#pragma unroll
    for (int k = 0; k < 4; ++k) {
      o[k] = o0[k]; o[k + 4] = o1[k];
      t[k] = t0[k]; t[k + 4] = t1[k];
    }
  }
#endif

  // Pass 1: row max for numerically stable softmax (both tensors at once).
  float lmo = o[0], lmt = t[0];
#pragma unroll
  for (int k = 1; k < 8; ++k) { lmo = fmaxf(lmo, o[k]); lmt = fmaxf(lmt, t[k]); }
  float2 mx = breduce2(lmo, lmt, /*max*/true, sA, sB, bc, lane, wave);
  const float mo = mx.x, mt = mx.y;

  // Pass 2: sum of exp(x - max); keep exp(t - mt) for the label weights.
  float so = 0.f, st = 0.f;
  float et[8];
#pragma unroll
  for (int k = 0; k < 8; ++k) {
    so += expf(o[k] - mo);
    float e = expf(t[k] - mt);
    et[k] = e;
    st += e;
  }
  float2 sm = breduce2(so, st, /*max*/false, sA, sB, bc, lane, wave);
  const float log_so = logf(sm.x);
  const float log_st = logf(sm.y);
  const float inv_st = 1.0f / sm.y;

  // Pass 3: kl = label * (log_label - log_score), accumulate.
  float kl = 0.f;
#pragma unroll
  for (int k = 0; k < 8; ++k) {
    float log_label = (t[k] - mt) - log_st;
    float log_score = (o[k] - mo) - log_so;
    kl += (et[k] * inv_st) * (log_label - log_score);
  }
  float2 ks = breduce2(kl, 0.f, /*max*/false, sA, sB, bc, lane, wave);
  if (tid == 0) row_out[row] = ks.x * (1.0f / (float)BINS);
}

// Kernel 2: one wave32 per joint. mean over batch, sum over d, min over joints.
__global__ __launch_bounds__(NJ * 32)
void kl_final_kernel(const float* __restrict__ rows, float* __restrict__ out) {
  const int tid  = threadIdx.x;
  const int lane = tid & 31;
  const int j    = tid >> 5;          // joint id, 0..16
  __shared__ float jloss[NJ];

  float acc = 0.f;
  if (j < NJ) {
    for (int b = lane; b < NB; b += 32) {
      size_t r = ((size_t)b * NJ + j) * ND;
      acc += rows[r] + rows[r + 1];   // sum over d
    }
#pragma unroll
    for (int m = 16; m >= 1; m >>= 1) acc += __shfl_xor(acc, m, 32);
    if (lane == 0) jloss[j] = acc * (1.0f / (float)NB);   // mean over batch
  }
  __syncthreads();
  if (tid == 0) {
    float mn = jloss[0];
#pragma unroll
    for (int jj = 1; jj < NJ; ++jj) mn = fminf(mn, jloss[jj]);
    out[0] = mn;
  }
}

extern "C" void kernel_launch(void* const* d_in, const int* in_sizes, int n_in,
                              void* d_out, int out_size, void* d_ws, size_t ws_size,
                              hipStream_t stream) {
  const float* pred = (const float*)d_in[0];   // "output" [256,17,2,2048] f32
  const float* tgt  = (const float*)d_in[1];   // "target" [256,17,2,2048] f32
  float* row_ws = (float*)d_ws;                // ROWS floats of scratch
  float* out    = (float*)d_out;               // 1 float

  kl_rows_kernel<<<ROWS, TPB, 0, stream>>>(pred, tgt, row_ws);
  kl_final_kernel<<<1, NJ * 32, 0, stream>>>(row_ws, out);
}